// FasterRCNN_89876485636849
// MI455X (gfx1250) — compile-verified
//
#include <hip/hip_runtime.h>

// Class-specific greedy NMS for N=8192 boxes, 80 classes, IoU thr = 0.5.
// Outputs (float): keep[8192] as 0/1, then kept_scores[8192].

#define N_BOXES 8192
#define MASK_W  (N_BOXES / 32)   // 256 mask words per row
#define THR     0.5f

typedef __attribute__((ext_vector_type(2))) float v2f;
typedef __attribute__((ext_vector_type(8))) float v8f;

// ---------------------------------------------------------------- K1: max coord
__global__ __launch_bounds__(1024) void nms_max_kernel(const float* __restrict__ boxes,
                                                       float* __restrict__ maxc) {
  __shared__ float sm[1024];
  float m = -3.0e38f;
  for (int i = threadIdx.x; i < N_BOXES * 4; i += 1024) m = fmaxf(m, boxes[i]);
  sm[threadIdx.x] = m;
  __syncthreads();
  for (int s = 512; s > 0; s >>= 1) {
    if (threadIdx.x < (unsigned)s) sm[threadIdx.x] = fmaxf(sm[threadIdx.x], sm[threadIdx.x + s]);
    __syncthreads();
  }
  if (threadIdx.x == 0) maxc[0] = sm[0];
}

// ------------------------------------------------- K2: class offset + box areas
__global__ __launch_bounds__(256) void nms_prep_kernel(const float* __restrict__ boxes,
                                                       const int* __restrict__ cls,
                                                       const float* __restrict__ maxc,
                                                       float* __restrict__ b,
                                                       float* __restrict__ area) {
  int i = blockIdx.x * blockDim.x + threadIdx.x;
  if (i >= N_BOXES) return;
  float off = (float)cls[i] * (maxc[0] + 1.0f);
  float x1 = boxes[i * 4 + 0] + off;
  float y1 = boxes[i * 4 + 1] + off;
  float x2 = boxes[i * 4 + 2] + off;
  float y2 = boxes[i * 4 + 3] + off;
  b[i * 4 + 0] = x1;
  b[i * 4 + 1] = y1;
  b[i * 4 + 2] = x2;
  b[i * 4 + 3] = y2;
  area[i] = (x2 - x1) * (y2 - y1);  // offsets cancel in the differences
}

// --------------------- K3: bitonic sort (desc score) wholly in LDS, then gather
__global__ __launch_bounds__(1024) void nms_sort_kernel(const float* __restrict__ scores,
                                                        const float* __restrict__ b,
                                                        const float* __restrict__ area,
                                                        float* __restrict__ bs,
                                                        float* __restrict__ as_,
                                                        int* __restrict__ sidx) {
  __shared__ float key[N_BOXES];  // 32 KB
  __shared__ int   val[N_BOXES];  // 32 KB  (total 64 KB << 320 KB WGP LDS)
  const int tid = threadIdx.x;
  for (int i = tid; i < N_BOXES; i += 1024) { key[i] = scores[i]; val[i] = i; }
  __syncthreads();
  for (int k = 2; k <= N_BOXES; k <<= 1) {
    for (int j = k >> 1; j > 0; j >>= 1) {
      for (int i = tid; i < N_BOXES; i += 1024) {
        int l = i ^ j;
        if (l > i) {
          float ki = key[i], kl = key[l];
          // overall DESCENDING sort
          bool sw = ((i & k) == 0) ? (ki < kl) : (ki > kl);
          if (sw) {
            key[i] = kl; key[l] = ki;
            int vi = val[i]; val[i] = val[l]; val[l] = vi;
          }
        }
      }
      __syncthreads();
    }
  }
  for (int r = tid; r < N_BOXES; r += 1024) {
    int o = val[r];
    sidx[r] = o;
    bs[r * 4 + 0] = b[o * 4 + 0];
    bs[r * 4 + 1] = b[o * 4 + 1];
    bs[r * 4 + 2] = b[o * 4 + 2];
    bs[r * 4 + 3] = b[o * 4 + 3];
    as_[r] = area[o];
  }
}

// -------- K4: suppression bitmask. Each wave32 does a 16-row x 32-col tile.
// Denominator term a1[i]+a2[j] via V_WMMA_F32_16X16X4_F32:
//   A(16x4): col0 = a1 (row areas), col1 = 1, cols 2-3 = 0
//   B(4x16): row0 = 1, row1 = a2 (col areas), rows 2-3 = 0
//   D[i][j] = a1[i] + a2[j]
// VGPR layouts per ISA 7.12.2 (32-bit A 16x4: lanes 0-15 hold K=0,1; lanes
// 16-31 hold K=2,3; C/D: lane<16 -> (M=v, N=lane), lane>=16 -> (M=v+8, N=lane-16)).
__global__ __launch_bounds__(256) void nms_mask_kernel(const float4* __restrict__ bs,
                                                       const float* __restrict__ as_,
                                                       unsigned* __restrict__ mask) {
  __shared__ float4 rbox[16];
  __shared__ float  ra[16];
  const int tid = threadIdx.x;
  const int r0 = blockIdx.y * 16;
  if (tid < 16) {
    rbox[tid] = bs[r0 + tid];
    ra[tid] = as_[r0 + tid];
  }
  __syncthreads();

  const int wave = tid >> 5;
  const int lane = tid & 31;
  const int wc = blockIdx.x * 8 + wave;        // mask word column [0,256)
  const int ca = wc * 32 + (lane & 15);        // column of subtile 0
  const int cb = ca + 16;                      // column of subtile 1

  const float4 cA = bs[ca]; const float aA = as_[ca];
  const float4 cB = bs[cb]; const float aB = as_[cb];
  const bool lo = lane < 16;

  v2f A;  A[0]  = lo ? ra[lane & 15] : 0.0f;  A[1]  = lo ? 1.0f : 0.0f;
  v2f B0; B0[0] = lo ? 1.0f : 0.0f;           B0[1] = lo ? aA : 0.0f;
  v2f B1; B1[0] = lo ? 1.0f : 0.0f;           B1[1] = lo ? aB : 0.0f;
  v8f z = {};
  v8f S0 = __builtin_amdgcn_wmma_f32_16x16x4_f32(false, A, false, B0, (short)0, z, false, false);
  v8f S1 = __builtin_amdgcn_wmma_f32_16x16x4_f32(false, A, false, B1, (short)0, z, false, false);
  // Pin the accumulators: opaque to the optimizer so the WMMAs are issued
  // exactly once per tile instead of being rematerialized per D-element use.
  asm volatile("" : "+v"(S0), "+v"(S1));

  const float t = THR, tp1 = 1.0f + THR;  // iou > t  <=>  inter*(1+t) > t*(a1+a2)
  unsigned balA[8], balB[8];
#pragma unroll
  for (int v = 0; v < 8; ++v) {
    const int M = lo ? v : (v + 8);
    const float4 rb = rbox[M];

    float iw = fmaxf(fminf(rb.z, cA.z) - fmaxf(rb.x, cA.x), 0.0f);
    float ih = fmaxf(fminf(rb.w, cA.w) - fmaxf(rb.y, cA.y), 0.0f);
    bool sA = (iw * ih) * tp1 > t * S0[v];

    iw = fmaxf(fminf(rb.z, cB.z) - fmaxf(rb.x, cB.x), 0.0f);
    ih = fmaxf(fminf(rb.w, cB.w) - fmaxf(rb.y, cB.y), 0.0f);
    bool sB = (iw * ih) * tp1 > t * S1[v];

    balA[v] = __builtin_amdgcn_ballot_w32(sA);  // lo 16: row v; hi 16: row v+8 (cols 0-15)
    balB[v] = __builtin_amdgcn_ballot_w32(sB);  // same rows, cols 16-31
  }

  // Lane-parallel mask store: lanes 0-15 each own one row's word (row = r0+lane):
  //   lanes 0-7  -> rows v   = lane   : low halves of balA/balB
  //   lanes 8-15 -> rows v+8 = lane   : high halves of balA/balB
  if (lane < 16) {
    unsigned a = 0u, bw = 0u;
    const int s = lane & 7;
#pragma unroll
    for (int v = 0; v < 8; ++v) {
      if (s == v) { a = balA[v]; bw = balB[v]; }  // uniform values, per-lane select
    }
    const unsigned word = (lane < 8)
        ? ((a & 0xFFFFu) | ((bw & 0xFFFFu) << 16))
        : ((a >> 16) | (bw & 0xFFFF0000u));
    mask[(size_t)(r0 + lane) * MASK_W + wc] = word;
  }
}

// ----------------- K5: sequential greedy scan with LDS bitset, scatter outputs
__global__ __launch_bounds__(256) void nms_scan_kernel(const unsigned* __restrict__ mask,
                                                       const int* __restrict__ sidx,
                                                       const float* __restrict__ scores,
                                                       float* __restrict__ out_keep,
                                                       float* __restrict__ out_score) {
  __shared__ unsigned removed[MASK_W];
  const int tid = threadIdx.x;
  removed[tid] = 0u;
  __syncthreads();
  for (int i = 0; i < N_BOXES; ++i) {
    const bool kept = ((removed[i >> 5] >> (i & 31)) & 1u) == 0u;  // uniform
    __syncthreads();
    if (kept) removed[tid] |= mask[(size_t)i * MASK_W + tid];
    if (tid == 0) {
      const int orig = sidx[i];
      out_keep[orig]  = kept ? 1.0f : 0.0f;
      out_score[orig] = kept ? scores[orig] : 0.0f;
    }
    __syncthreads();
  }
}

// --------------------------------------------------------------------- launch
extern "C" void kernel_launch(void* const* d_in, const int* in_sizes, int n_in,
                              void* d_out, int out_size, void* d_ws, size_t ws_size,
                              hipStream_t stream) {
  const float* boxes  = (const float*)d_in[0];
  const float* scores = (const float*)d_in[1];
  const int*   cls    = (const int*)d_in[2];

  // Workspace partition (all chunks multiples of 16 B; hipMalloc base is aligned)
  float* maxc = (float*)d_ws;                 // 4 floats
  float* b    = maxc + 4;                     // 8192*4 offset boxes (orig order)
  float* area = b + N_BOXES * 4;              // 8192 areas (orig order)
  float* bs   = area + N_BOXES;               // 8192*4 offset boxes (sorted)
  float* as_  = bs + N_BOXES * 4;             // 8192 areas (sorted)
  int*   sidx = (int*)(as_ + N_BOXES);        // 8192 sorted->orig index
  unsigned* mask = (unsigned*)(sidx + N_BOXES); // 8192*256 words = 8 MB

  float* out_keep  = (float*)d_out;
  float* out_score = out_keep + N_BOXES;

  hipLaunchKernelGGL(nms_max_kernel,  dim3(1),              dim3(1024), 0, stream, boxes, maxc);
  hipLaunchKernelGGL(nms_prep_kernel, dim3(N_BOXES / 256),  dim3(256),  0, stream, boxes, cls, maxc, b, area);
  hipLaunchKernelGGL(nms_sort_kernel, dim3(1),              dim3(1024), 0, stream, scores, b, area, bs, as_, sidx);
  hipLaunchKernelGGL(nms_mask_kernel, dim3(MASK_W / 8, N_BOXES / 16), dim3(256), 0, stream,
                     (const float4*)bs, as_, mask);
  hipLaunchKernelGGL(nms_scan_kernel, dim3(1),              dim3(256),  0, stream,
                     mask, sidx, scores, out_keep, out_score);
}